// LightGCN_84670985273432
// MI455X (gfx1250) — compile-verified
//
#include <hip/hip_runtime.h>
#include <stdint.h>

#define NUSERS  100000
#define NNODES  150000
#define DIM     64
#define NLAYERS 3

// ---------------------------------------------------------------------------
// init: cur = E0 = [user;item], acc(d_out) = E0, nxt = 0   (float4 streams)
// ---------------------------------------------------------------------------
__global__ __launch_bounds__(256) void lgcn_init(
    const float4* __restrict__ user4, const float4* __restrict__ item4,
    float4* __restrict__ cur4, float4* __restrict__ acc4,
    float4* __restrict__ nxt4, int total4, int user_total4) {
  int i = blockIdx.x * blockDim.x + threadIdx.x;
  if (i >= total4) return;
  float4 v = (i < user_total4) ? user4[i] : item4[i - user_total4];
  cur4[i] = v;
  acc4[i] = v;
  nxt4[i] = make_float4(0.f, 0.f, 0.f, 0.f);
}

// ---------------------------------------------------------------------------
// SpMM: nxt[row,:] += val * cur[col,:]  for each COO edge.
// One wave handles 2 edges per iteration with a 2-deep async pipeline:
//   gather : lanes 0-15 -> edge0 row, lanes 16-31 -> edge1 row, 16B/lane via
//            global_load_async_to_lds_b128 into one of two 512B LDS slots;
//            next pair is prefetched while the current one is consumed
//            (s_wait_asynccnt 1 -> older load complete, in-order rule)
//   scatter: all 32 lanes per edge, 2 dims/lane, global_atomic_add_f32
// ---------------------------------------------------------------------------
__global__ __launch_bounds__(256) void lgcn_spmm(
    const int* __restrict__ erow, const int* __restrict__ ecol,
    const float* __restrict__ eval, const float* __restrict__ cur,
    float* __restrict__ nxt, int n_edges) {
  __shared__ float smem[8 * 256];  // 8 waves/block * 2 slots * 128 f32 (8KB)
  const int lane   = threadIdx.x & 31;
  const int wv     = threadIdx.x >> 5;
  const int gwave  = blockIdx.x * (blockDim.x >> 5) + wv;
  const int nwaves = gridDim.x * (blockDim.x >> 5);
  const int n_pairs = (n_edges + 1) >> 1;

  const int pick = lane >> 4;   // which of the two edges this lane gathers
  const int sub  = lane & 15;   // 16B chunk within the 256B row
  float* wbuf = &smem[wv * 256];
  // low 32 bits of the generic LDS pointer == byte offset in our LDS alloc
  const uint32_t lds_lane =
      (uint32_t)(uintptr_t)(wbuf) + (uint32_t)(pick * 256 + sub * 16);

  // issue the async gather of both rows of edge-pair p into LDS slot `slot`
  auto issue = [&](int p, int slot) {
    int eg = (p << 1) + pick;
    if (eg > n_edges - 1) eg = n_edges - 1;  // clamp -> always valid address
    const int col = ecol[eg];
    const uint64_t gaddr =
        (uint64_t)(uintptr_t)(cur + (size_t)col * DIM + (size_t)(sub * 4));
    const uint32_t ldsoff = lds_lane + (uint32_t)(slot << 9);  // +0 / +512B
    asm volatile("global_load_async_to_lds_b128 %0, %1, off"
                 :: "v"(ldsoff), "v"(gaddr) : "memory");
  };

  int p = gwave;
  if (p >= n_pairs) return;          // wave-uniform
  issue(p, 0);                       // prologue prefetch
  int buf = 0;

  for (; p < n_pairs; p += nwaves) {
    const int pn = p + nwaves;       // wave-uniform
    if (pn < n_pairs) {
      issue(pn, buf ^ 1);            // prefetch next pair into other slot
      asm volatile("s_wait_asynccnt 0x1" ::: "memory");  // current slot ready
    } else {
      asm volatile("s_wait_asynccnt 0x0" ::: "memory");  // drain (epilogue)
    }

    const float* rbuf = wbuf + buf * 128;
    const int e0 = p << 1;
#pragma unroll
    for (int k = 0; k < 2; ++k) {
      const int e = e0 + k;          // wave-uniform bound check
      if (e < n_edges) {
        const int   row = erow[e];
        const float val = eval[e];
        const float2 f = *(const float2*)&rbuf[k * 64 + lane * 2];
        const float a0 = val * f.x;
        const float a1 = val * f.y;
        const uint64_t oaddr =
            (uint64_t)(uintptr_t)(nxt + (size_t)row * DIM + (size_t)(lane * 2));
        asm volatile("global_atomic_add_f32 %0, %1, off"
                     :: "v"(oaddr), "v"(a0) : "memory");
        asm volatile("global_atomic_add_f32 %0, %1, off offset:4"
                     :: "v"(oaddr), "v"(a1) : "memory");
      }
    }
    buf ^= 1;
  }
}

// ---------------------------------------------------------------------------
// accum: acc = (acc + nxt) * scale ; optionally zero the buffer that becomes
// the next scatter target (skipped on the final layer: saves a 38MB stream)
// ---------------------------------------------------------------------------
__global__ __launch_bounds__(256) void lgcn_accum(
    float4* __restrict__ acc4, const float4* __restrict__ nxt4,
    float4* __restrict__ zer4, int total4, float scale, int do_zero) {
  int i = blockIdx.x * blockDim.x + threadIdx.x;
  if (i >= total4) return;
  float4 a = acc4[i];
  float4 n = nxt4[i];
  a.x = (a.x + n.x) * scale;
  a.y = (a.y + n.y) * scale;
  a.z = (a.z + n.z) * scale;
  a.w = (a.w + n.w) * scale;
  acc4[i] = a;
  if (do_zero) zer4[i] = make_float4(0.f, 0.f, 0.f, 0.f);
}

// ---------------------------------------------------------------------------
extern "C" void kernel_launch(void* const* d_in, const int* in_sizes, int n_in,
                              void* d_out, int out_size, void* d_ws, size_t ws_size,
                              hipStream_t stream) {
  const float* user = (const float*)d_in[0];
  const float* item = (const float*)d_in[1];
  const int*   erow = (const int*)d_in[2];
  const int*   ecol = (const int*)d_in[3];
  const float* eval = (const float*)d_in[4];
  const int n_edges = in_sizes[2];

  float* acc = (float*)d_out;                    // 150000 x 64 f32
  float* B0  = (float*)d_ws;                     // ping
  float* B1  = B0 + (size_t)NNODES * DIM;        // pong (needs ~76.8MB ws)

  const int total4      = NNODES * DIM / 4;      // 2.4M float4
  const int user_total4 = NUSERS * DIM / 4;
  dim3 blk(256);
  dim3 grid_n((total4 + 255) / 256);

  lgcn_init<<<grid_n, blk, 0, stream>>>((const float4*)user, (const float4*)item,
                                        (float4*)B0, (float4*)acc, (float4*)B1,
                                        total4, user_total4);

  float* cur = B0;
  float* nxt = B1;
  const int spmm_blocks = 3072;  // 24576 waves; grid-stride over 500K edge-pairs
  for (int l = 0; l < NLAYERS; ++l) {
    lgcn_spmm<<<spmm_blocks, blk, 0, stream>>>(erow, ecol, eval, cur, nxt, n_edges);
    const int   last  = (l == NLAYERS - 1);
    const float scale = last ? 0.25f : 1.0f;
    lgcn_accum<<<grid_n, blk, 0, stream>>>((float4*)acc, (const float4*)nxt,
                                           (float4*)cur, total4, scale, !last);
    float* t = cur; cur = nxt; nxt = t;
  }
  (void)n_in; (void)out_size; (void)ws_size;
}